// CoExCostVolume_33328946217106
// MI455X (gfx1250) — compile-verified
//
#include <hip/hip_runtime.h>

#define B_    8
#define C_    96
#define H_    128
#define W_    416
#define D_    49
#define HW_   (H_ * W_)
#define PAIRS 48            // channel pairs (c/2)
#define GUARD 96            // dwords of zeros per pair-row (48 cols x 2 ch) -> branchless zero-fill
#define PSTR  928           // pair-row stride in dwords: 928 % 64 == 32 -> lane halves on disjoint banks
#define WB    104           // 16B w-blocks per row (416/4)
#define TSTR  17            // tile-buffer stride: gcd(17,64)==1 -> conflict-free scatter
#define NT    416           // 13 wave32 -> exactly 2 w-tiles per wave
#define NWAVE 13

typedef __attribute__((ext_vector_type(2))) float v2f;
typedef __attribute__((ext_vector_type(4))) float v4f;
typedef __attribute__((ext_vector_type(8))) float v8f;
typedef __attribute__((ext_vector_type(4))) int   v4i;

#if defined(__HIP_DEVICE_COMPILE__)
#if !__has_builtin(__builtin_amdgcn_wmma_f32_16x16x4_f32)
#error "missing __builtin_amdgcn_wmma_f32_16x16x4_f32 on gfx1250 device pass"
#endif
#if __has_builtin(__builtin_amdgcn_global_load_async_to_lds_b128)
#define HAVE_ASYNC_LDS 1
#endif
#endif

__global__ __launch_bounds__(NT) void costvol_wmma_f32(const float* __restrict__ x,
                                                       const float* __restrict__ y,
                                                       float* __restrict__ out,
                                                       const float* __restrict__ zeros) {
  extern __shared__ float lds[];
  float* ylds = lds;                   // PAIRS * PSTR dwords: guard + pair-interleaved y row
  float* tbuf = lds + PAIRS * PSTR;    // NWAVE * D_ * TSTR floats: transpose buffers

  const int tid = threadIdx.x;
  const int bh  = blockIdx.x;
  const int b   = bh / H_;
  const int h   = bh - b * H_;

  const float* yrow = y + (size_t)b * C_ * HW_ + (size_t)h * W_;

  // ---- guard zones: DMA 18KB of zeros into LDS via async-to-LDS (src = memset d_ws) ----
#if defined(HAVE_ASYNC_LDS)
  for (int i = tid; i < PAIRS * (GUARD / 4); i += NT) {     // 1152 x 16B blocks
    int p = i / (GUARD / 4);
    int q = i - p * (GUARD / 4);
    __builtin_amdgcn_global_load_async_to_lds_b128(
        (v4i*)(zeros + 4 * (i & 255)),                      // stays inside 4KB zero pool
        (v4i*)&ylds[p * PSTR + 4 * q], 0, 0);
  }
  asm volatile("s_wait_asynccnt 0x0" ::: "memory");
#else
  (void)zeros;
  for (int i = tid; i < PAIRS * (GUARD / 4); i += NT) {
    int p = i / (GUARD / 4);
    int q = i - p * (GUARD / 4);
    v4f z = {};
    *(v4f*)&ylds[p * PSTR + 4 * q] = z;
  }
#endif

  // ---- stage y row channel-pair-interleaved: [y[2p][w], y[2p+1][w]] at consecutive dwords ----
  for (int u = tid; u < PAIRS * WB; u += NT) {              // 4992/416 = 12 exact
    int p = u / WB;
    int q = u - p * WB;
    const float* ga = yrow + (size_t)(2 * p) * HW_ + 4 * q;
    v4f ya = *(const v4f*)ga;
    v4f yb = *(const v4f*)(ga + HW_);
    float* d = &ylds[p * PSTR + GUARD + 8 * q];
    v4f lo = {ya.x, yb.x, ya.y, yb.y};
    v4f hi = {ya.z, yb.z, ya.w, yb.w};
    *(v4f*)d       = lo;
    *(v4f*)(d + 4) = hi;
  }
  __syncthreads();

  const int wave  = tid >> 5;
  const int lane  = tid & 31;
  const int lhalf = lane >> 4;   // 0 | 1
  const int lmod  = lane & 15;
  float* myt = tbuf + wave * (D_ * TSTR);

  const float* xrow = x + (size_t)b * C_ * HW_ + (size_t)h * W_;

  // Exactly two 16-wide w-tiles per wave: wt = wave and wave + 13
#pragma unroll
  for (int t = 0; t < 2; ++t) {
    const int wt = wave + NWAVE * t;
    const int w0 = wt * 16;
    v8f acc0 = {}, acc1 = {}, acc2 = {}, acc3 = {};
    const float* xp = xrow + w0 + lmod;

    for (int k = 0; k < C_; k += 4) {
      // A fragment (16x4 f32): lane half 0 -> K = k,k+1 ; half 1 -> K = k+2,k+3
      const int ca = k + lhalf * 2;
      v2f a;
      a.x = xp[(size_t)ca * HW_];
      a.y = xp[(size_t)(ca + 1) * HW_];
      __builtin_prefetch(xp + (size_t)(ca + 8) * HW_, 0, 1);

      // B fragments: one aligned ds_load_b64 each -> {row ca, row ca+1} already paired
      const int p = (k >> 1) + lhalf;                        // pair-row holding (ca, ca+1)
      const float* yb = ylds + p * PSTR + GUARD + 2 * (w0 + lmod);
      v2f b0 = *(const v2f*)(yb);        // j=0: cols w0..w0+15
      v2f b1 = *(const v2f*)(yb - 32);   // j=1: cols w0-16..
      v2f b2 = *(const v2f*)(yb - 64);   // j=2: cols w0-32..
      v2f b3 = *(const v2f*)(yb - 96);   // j=3: cols w0-48.. (guard zone supplies zeros)

      // G_j += X_tile^T * Y_tile(w0 - 16j), fp32 WMMA, K += 4 per step
      acc0 = __builtin_amdgcn_wmma_f32_16x16x4_f32(false, a, false, b0, (short)0, acc0, false, false);
      acc1 = __builtin_amdgcn_wmma_f32_16x16x4_f32(false, a, false, b1, (short)0, acc1, false, false);
      acc2 = __builtin_amdgcn_wmma_f32_16x16x4_f32(false, a, false, b2, (short)0, acc2, false, false);
      acc3 = __builtin_amdgcn_wmma_f32_16x16x4_f32(false, a, false, b3, (short)0, acc3, false, false);
    }

    // ---- Hankel remap: cost[d][w] = G_j[M][N] with d = M + 16j - N ----
    // C/D layout: VGPR i -> M = i + 8*lhalf, N = lmod. Each (M,d) hits exactly one (j,N).
#pragma unroll
    for (int i = 0; i < 8; ++i) {
      const int M  = i + 8 * lhalf;
      const int d0 = M - lmod;                          // j=0 disparity
      if (d0 >= 0) myt[d0 * TSTR + M] = acc0[i];        // d in [0, M]
      myt[(d0 + 16) * TSTR + M] = acc1[i];              // d in [M+1, M+16]
      myt[(d0 + 32) * TSTR + M] = acc2[i];              // d in [M+17, M+32]
      if (d0 <= 0) myt[(d0 + 48) * TSTR + M] = acc3[i]; // d in [M+33, 48]
    }

    // ---- Coalesced store: two disparity rows per pass, 64B per half-wave ----
    float* obase = out + ((size_t)b * D_ * H_ + h) * (size_t)W_ + w0 + lmod;
    for (int r = lhalf; r < D_; r += 2) {
      obase[(size_t)r * HW_] = myt[r * TSTR + lmod];
    }
  }
}

extern "C" void kernel_launch(void* const* d_in, const int* in_sizes, int n_in,
                              void* d_out, int out_size, void* d_ws, size_t ws_size,
                              hipStream_t stream) {
  (void)in_sizes; (void)n_in; (void)out_size; (void)ws_size;
  const float* x = (const float*)d_in[0];
  const float* y = (const float*)d_in[1];
  float* out     = (float*)d_out;

  // 4KB zero pool for the async guard-zone fill (graph-capturable memset node)
  hipMemsetAsync(d_ws, 0, 4096, stream);

  const dim3 grid(B_ * H_);    // one workgroup per (b, h) row
  const dim3 block(NT);        // 13 wave32 -> 2 tiles per wave, perfectly balanced
  const size_t shmem = (size_t)(PAIRS * PSTR + NWAVE * D_ * TSTR) * sizeof(float); // ~216 KB

  costvol_wmma_f32<<<grid, block, shmem, stream>>>(x, y, out, (const float*)d_ws);
}